// Norm_6725918785724
// MI455X (gfx1250) — compile-verified
//
#include <hip/hip_runtime.h>
#include <hip/hip_bf16.h>
#include <stdint.h>

// ---------------------------------------------------------------------------
// GraphNorm, fused per-segment (batch is sorted -> segments are contiguous).
//   pass A: per-graph sum / sumsq via GLOBAL_LOAD_ASYNC_TO_LDS_B128 streaming
//   pass B: out = x*scale + shift, re-read hits L2 (192MB), NT load/store
// Roofline: ~1.0-1.5 GB HBM traffic, FLOPs negligible -> pure streaming kernel.
// ---------------------------------------------------------------------------

#define EPSV 1e-9f
constexpr int H    = 256;           // hidden (fixed by reference)
constexpr int H4   = H / 4;         // float4s per row
constexpr int GNUM = 4096;          // num_graphs (fixed by reference)
constexpr int TR   = 16;            // tile rows per async stage (16 KB)
constexpr int TQ   = TR * H4;       // float4s per tile = 1024
constexpr int PT   = TQ / 256;      // float4s per thread per tile = 4

typedef float v4f __attribute__((ext_vector_type(4)));
typedef int   v4i __attribute__((ext_vector_type(4)));

// ---- CDNA5 async-to-LDS path (builtin confirmed present by probe round) ---
#if defined(__has_builtin)
#if __has_builtin(__builtin_amdgcn_global_load_async_to_lds_b128)
#define HAVE_ASYNC_BUILTIN 1
#endif
#if __has_builtin(__builtin_amdgcn_s_wait_asynccnt)
#define WAIT_ASYNC(n) __builtin_amdgcn_s_wait_asynccnt(n)
#endif
#endif
#ifndef WAIT_ASYNC
#define WAIT_ASYNC(n) asm volatile("s_wait_asynccnt %0" ::"i"(n) : "memory")
#endif

// Exact parameter types per clang diagnostic: pointer-to-v4i in AS1 / AS3.
typedef __attribute__((address_space(1))) v4i* gas_v4i;
typedef __attribute__((address_space(3))) v4i* las_v4i;

__device__ __forceinline__ void async_cp16(const v4f* gsrc, const v4f* ldst) {
#ifdef HAVE_ASYNC_BUILTIN
  __builtin_amdgcn_global_load_async_to_lds_b128(
      (gas_v4i)(uintptr_t)gsrc,
      (las_v4i)(uint32_t)(uintptr_t)ldst,  // low 32 bits = LDS offset
      /*offset=*/0, /*cpol=*/0);
#else
  asm volatile("global_load_async_to_lds_b128 %0, %1, off"
               :: "v"((uint32_t)(uintptr_t)ldst),
                  "v"((uint64_t)(uintptr_t)gsrc)
               : "memory");
#endif
}

// ---------------------------------------------------------------------------
// Kernel 1: segment starts from sorted batch ids. seg[g] = first n with
// batch[n] >= g, for g in [0, G]; seg[G] = N.
// ---------------------------------------------------------------------------
__global__ void gn_seg_kernel(const int* __restrict__ batch,
                              int* __restrict__ seg, int N, int G) {
  int n = blockIdx.x * blockDim.x + threadIdx.x;
  if (n >= N) return;
  int b    = batch[n];
  int prev = (n == 0) ? -1 : batch[n - 1];
  for (int g = prev + 1; g <= b; ++g) seg[g] = n;
  if (n == N - 1)
    for (int g = b + 1; g <= G; ++g) seg[g] = N;
}

// ---------------------------------------------------------------------------
// Kernel 2: one 256-thread block (8 wave32) per graph.
// Lane mapping (both passes): channel-quad c4 = t&63, row-group tg = t>>6.
// Each lane consumes exactly the LDS float4s it DMA'd -> no cross-wave sync
// in the async pipeline; per-wave s_wait_asynccnt is sufficient.
// ---------------------------------------------------------------------------
__device__ __forceinline__ void gn_issue_tile(const v4f* __restrict__ x4,
                                              int s, int rows, int it,
                                              const v4f* buf, int t) {
#pragma unroll
  for (int j = 0; j < PT; ++j) {
    int q = t + j * 256;            // flat float4 index within tile
    int r = it * TR + (q >> 6);     // row within segment
    if (r >= rows) r = rows - 1;    // clamp tail (accumulation is guarded)
    long f = (long)(s + r) * H4 + (q & 63);
    async_cp16(&x4[f], &buf[q]);
  }
}

__global__ __launch_bounds__(256)
void gn_fused_kernel(const float* __restrict__ x, const int* __restrict__ seg,
                     const float* __restrict__ alpha,
                     const float* __restrict__ weight,
                     const float* __restrict__ bias,
                     float* __restrict__ out, int N) {
  __shared__ v4f buf[2][TQ];        // 2 x 16 KB double buffer
  __shared__ v4f red[256];          // cross-rowgroup reduction
  __shared__ v4f s_scale[H4];
  __shared__ v4f s_shift[H4];

  const int g  = blockIdx.x;
  const int s  = seg[g];
  const int e  = seg[g + 1];
  const int rows = e - s;
  const int t  = threadIdx.x;
  const int tg = t >> 6;            // row group 0..3
  const int c4 = t & 63;            // channel quad 0..63

  const v4f* x4 = (const v4f*)x;
  v4f acc_s = (v4f)0.0f;
  v4f acc_q = (v4f)0.0f;

  // ---------------- pass A: streaming sum / sumsq via async-to-LDS ---------
  if (rows > 0) {
    const int ntiles = (rows + TR - 1) / TR;
    gn_issue_tile(x4, s, rows, 0, buf[0], t);
    for (int it = 0; it < ntiles; ++it) {
      if (it + 1 < ntiles) {
        gn_issue_tile(x4, s, rows, it + 1, buf[(it + 1) & 1], t);
        WAIT_ASYNC(PT);             // oldest tile's PT loads complete
      } else {
        WAIT_ASYNC(0);
      }
      const v4f* b = buf[it & 1];
#pragma unroll
      for (int j = 0; j < PT; ++j) {
        int q = t + j * 256;
        int r = it * TR + (q >> 6);
        v4f v = b[q];               // ds_load_b128
        if (r < rows) {
          acc_s += v;
          acc_q += v * v;
        }
      }
    }
  }

  // ---------------- reduce 4 row-groups, compute scale/shift ---------------
  red[t] = acc_s;
  __syncthreads();
  v4f rs = (v4f)0.0f, rq = (v4f)0.0f;
  if (tg == 0) rs = red[t] + red[t + 64] + red[t + 128] + red[t + 192];
  __syncthreads();
  red[t] = acc_q;
  __syncthreads();
  if (tg == 0) {
    rq = red[t] + red[t + 64] + red[t + 128] + red[t + 192];
    float inv = 1.0f / (float)(rows > 0 ? rows : 1);
    v4f a  = ((const v4f*)alpha)[c4];
    v4f w  = ((const v4f*)weight)[c4];
    v4f bz = ((const v4f*)bias)[c4];
    v4f mean = rs * inv;
    v4f ex2  = rq * inv;
    // var = E[x^2] - 2a*mean^2 + a^2*mean^2 + eps  (variance of x - a*mean)
    v4f var = ex2 + a * (a - 2.0f) * (mean * mean) + EPSV;
    v4f sc;
    sc.x = w.x / sqrtf(var.x);
    sc.y = w.y / sqrtf(var.y);
    sc.z = w.z / sqrtf(var.z);
    sc.w = w.w / sqrtf(var.w);
    v4f sh = bz - a * mean * sc;
    s_scale[c4] = sc;
    s_shift[c4] = sh;
  }
  __syncthreads();

  // ---------------- pass B: apply (rows are L2-hot), NT in / NT out --------
  v4f sc = s_scale[c4];
  v4f sh = s_shift[c4];
  v4f* o4 = (v4f*)out;
  for (int r = tg; r < rows; r += 4) {
    long f = (long)(s + r) * H4 + c4;
    v4f v = __builtin_nontemporal_load(&x4[f]);   // last use: don't retain
    v4f o = v * sc + sh;
    __builtin_nontemporal_store(o, &o4[f]);       // streaming store
  }
}

// ---------------------------------------------------------------------------
extern "C" void kernel_launch(void* const* d_in, const int* in_sizes, int n_in,
                              void* d_out, int out_size, void* d_ws,
                              size_t ws_size, hipStream_t stream) {
  const float* x      = (const float*)d_in[0];
  const int*   batch  = (const int*)d_in[1];
  const float* alpha  = (const float*)d_in[2];
  const float* weight = (const float*)d_in[3];
  const float* bias   = (const float*)d_in[4];
  float*       out    = (float*)d_out;

  const int N = in_sizes[1];        // node count (batch length)
  int* seg = (int*)d_ws;            // (GNUM + 1) ints of scratch

  gn_seg_kernel<<<(N + 255) / 256, 256, 0, stream>>>(batch, seg, N, GNUM);
  gn_fused_kernel<<<GNUM, 256, 0, stream>>>(x, seg, alpha, weight, bias, out, N);
}